// SpatioTemporalEmbLoss_36610301231305
// MI455X (gfx1250) — compile-verified
//
#include <hip/hip_runtime.h>
#include <hip/hip_bf16.h>

// ---------------------------------------------------------------------------
// SpatioTemporalEmbLoss for MI455X (gfx1250, wave32)
//
// Pipeline:
//   k_init      : zero small accumulator region in ws
//   k_stats     : per-(b,k) mask sums (count, t/y/x coord sums, sigma, sigma^2)
//                 + per-b seed_bg  (one streaming pass, LDS atomics + global)
//   k_finalize  : centers, |c|^2, s_exp=exp(10*s_mean), var_k, present, obj
//   k_distkeys  : WMMA f32 16x16x4 computes sqdist for 16 voxels x 16 inst
//                 columns per wave; emits radix keys (~bits(err) | label) and
//                 accumulates seed_fg.  Native v_tanh_f32 + v_rcp_f32 paths.
//   4x radix    : hist -> scan -> stable scatter (ballot-match wave ranking)
//   k_lcnt/lscan/k_lovasz : positional ones-scan + Lovasz hinge evaluation
//   k_final     : combine  mean_b(inst + 10*var + seed)
// ---------------------------------------------------------------------------

typedef __attribute__((ext_vector_type(2))) float v2f;
typedef __attribute__((ext_vector_type(8))) float v8f;

#define NVOX   (1 << 20)     // T*H*W = 8*256*512
#define NSEGS  16            // B*K_MAX
#define SBLKS  256           // radix blocks per segment
#define STILE  4096          // elements per radix block
#define DT_C   (0.1f  / 31.0f)
#define DY_C   (1.6f  / 799.0f)
#define DX_C   (4.16f / 1999.0f)

// v_rcp_f32 instead of the ~10-instruction IEEE divide chain
static __device__ __forceinline__ float fast_rcp(float x) {
    return __builtin_amdgcn_rcpf(x);
}
static __device__ __forceinline__ float sigmoidf_(float x) {
    return fast_rcp(1.0f + __expf(-x));
}
// CDNA5 native transcendental tanh; v_nop covers the TRANS-op result hazard
// (inline asm bypasses the compiler's hazard scheduler).
static __device__ __forceinline__ float fast_tanh(float x) {
    float r;
    asm volatile("v_tanh_f32 %0, %1\n\tv_nop" : "=v"(r) : "v"(x));
    return r;
}

// ---------------------------------------------------------------- init ------
__global__ void k_init(float* acc) {
    // zero 256 floats: stats(98) + derived(128) + obj(2) + instLoss(16) + seedfg(2)
    if (threadIdx.x < 256) acc[threadIdx.x] = 0.0f;
}

// --------------------------------------------------------------- stats ------
// stats layout: [b][k][6] = count,sumT,sumY,sumX,sumS,sumS2  (96) ; seed_bg[2] at 96..97
__global__ void k_stats(const float* __restrict__ pred, const int* __restrict__ inst,
                        float* __restrict__ stats) {
    __shared__ float acc[98];
    for (int i = threadIdx.x; i < 98; i += blockDim.x) acc[i] = 0.0f;
    __syncthreads();
    const int total = 2 * NVOX;
    for (int idx = blockIdx.x * blockDim.x + threadIdx.x; idx < total;
         idx += gridDim.x * blockDim.x) {
        const int b = idx >> 20;
        const int v = idx & (NVOX - 1);
        const int g = inst[idx];
        const float sig  = pred[((size_t)(b * 5 + 3) << 20) + v];
        const float sraw = pred[((size_t)(b * 5 + 4) << 20) + v];
        const float seed = sigmoidf_(sraw);
        if (g == 0) {
            atomicAdd(&acc[96 + b], seed * seed);
        } else if (g <= 8) {
            const int o  = (b * 8 + (g - 1)) * 6;
            const int ti = v >> 17, yi = (v >> 9) & 255, xi = v & 511;
            atomicAdd(&acc[o + 0], 1.0f);
            atomicAdd(&acc[o + 1], ti * DT_C);
            atomicAdd(&acc[o + 2], yi * DY_C);
            atomicAdd(&acc[o + 3], xi * DX_C);
            atomicAdd(&acc[o + 4], sig);
            atomicAdd(&acc[o + 5], sig * sig);
        }
    }
    __syncthreads();
    for (int i = threadIdx.x; i < 98; i += blockDim.x)
        if (acc[i] != 0.0f) atomicAdd(&stats[i], acc[i]);
}

// ------------------------------------------------------------ finalize ------
// derived layout: [b*8+k][8] = cx,cy,cz,c2,s_exp,present,var_k,count
__global__ void k_finalize(const float* __restrict__ stats, float* __restrict__ derived,
                           float* __restrict__ obj) {
    const int t = threadIdx.x;
    __shared__ float pres[16];
    if (t < 16) {
        const float* s   = stats + t * 6;
        const float cnt  = s[0];
        const float safe = fmaxf(cnt, 1.0f);
        const float cx = s[1] / safe, cy = s[2] / safe, cz = s[3] / safe;
        const float sm = s[4] / safe;
        const float vr = s[5] / safe - sm * sm;     // forward: s_sg == s_mean
        const float p  = (cnt > 0.0f) ? 1.0f : 0.0f;
        float* dp = derived + t * 8;
        dp[0] = cx; dp[1] = cy; dp[2] = cz;
        dp[3] = cx * cx + cy * cy + cz * cz;
        dp[4] = __expf(10.0f * sm);
        dp[5] = p; dp[6] = vr; dp[7] = cnt;
        pres[t] = p;
    }
    __syncthreads();
    if (t < 2) {
        float o = 0.0f;
        for (int k = 0; k < 8; ++k) o += pres[t * 8 + k];
        obj[t] = o;
    }
}

// ------------------------------------------------------------ distkeys ------
// One wave handles 16 voxels x 16 instance columns per WMMA f32 16x16x4.
//   A[m,:] = [sx, sy, sz, |s|^2]   (lanes 0-15: K=0,1 ; lanes 16-31: K=2,3)
//   B[:,n] = [-2cx, -2cy, -2cz, 1] (same split)
//   C[m,n] = |c_n|^2
//   D[m,n] = |s_m|^2 - 2 s_m.c_n + |c_n|^2 = sqdist
__global__ void __launch_bounds__(256)
k_distkeys(const float* __restrict__ pred, const int* __restrict__ inst,
           const float* __restrict__ derived, unsigned* __restrict__ keys,
           float* __restrict__ seedfg) {
    const int b    = blockIdx.y;
    const int lane = threadIdx.x & 31;
    const int wid  = threadIdx.x >> 5;
    const int n    = lane & 15;        // instance column (valid if n < 8)
    const int hs   = lane >> 4;        // half-wave select (A/B K split)

    float cx = 0.f, cy = 0.f, cz = 0.f, c2 = 0.f, sexp = 0.f;
    if (n < 8) {
        const float* dp = derived + (b * 8 + n) * 8;
        cx = dp[0]; cy = dp[1]; cz = dp[2]; c2 = dp[3]; sexp = dp[4];
    }
    v2f Bm;
    if (hs == 0) { Bm[0] = -2.0f * cx; Bm[1] = -2.0f * cy; }
    else         { Bm[0] = -2.0f * cz; Bm[1] = (n < 8) ? 1.0f : 0.0f; }

    const float* p0 = pred + ((size_t)(b * 5 + 0) << 20);
    const float* p1 = pred + ((size_t)(b * 5 + 1) << 20);
    const float* p2 = pred + ((size_t)(b * 5 + 2) << 20);
    const float* p4 = pred + ((size_t)(b * 5 + 4) << 20);
    const int*   ib = inst + ((size_t)b << 20);

    const int waveId = blockIdx.x * (blockDim.x >> 5) + wid;
    const int nWaves = gridDim.x * (blockDim.x >> 5);
    float localSeed = 0.0f;

    for (int tile = waveId; tile < NVOX / 16; tile += nWaves) {
        const int base = tile * 16;
        const int va   = base + n;
        __builtin_prefetch(p0 + va + 16 * nWaves, 0, 1);   // global_prefetch_b8

        // hoist instance-id loads so they overlap tanh + WMMA compute
        int gvals[8];
        if (n < 8) {
#pragma unroll
            for (int j = 0; j < 8; ++j) gvals[j] = ib[base + j + (hs << 3)];
        }

        const float q0 = p0[va], q1 = p1[va], q2 = p2[va];
        const int ti = va >> 17, yi = (va >> 9) & 255, xi = va & 511;
        const float sx = fast_tanh(q0) + ti * DT_C;
        const float sy = fast_tanh(q1) + yi * DY_C;
        const float sz = fast_tanh(q2) + xi * DX_C;
        const float e2 = sx * sx + sy * sy + sz * sz;

        v2f A;
        if (hs == 0) { A[0] = sx; A[1] = sy; }
        else         { A[0] = sz; A[1] = e2; }
        v8f C;
#pragma unroll
        for (int j = 0; j < 8; ++j) C[j] = c2;

        // uniform control flow here: EXEC all-ones for WMMA
        v8f D = __builtin_amdgcn_wmma_f32_16x16x4_f32(
            false, A, false, Bm, (short)0, C, false, false);

        if (n < 8) {
            const unsigned segBase = ((unsigned)(b * 8 + n)) << 20;
#pragma unroll
            for (int j = 0; j < 8; ++j) {
                const int m  = j + (hs << 3);
                const int vm = base + m;
                const float sq = fmaxf(D[j], 0.0f);
                const float d  = __expf(-sexp * sq);
                const unsigned l = (gvals[j] == n + 1) ? 1u : 0u;
                float e = l ? (2.0f - 2.0f * d) : (2.0f * d);
                e = fmaxf(e, 0.0f);
                // descending float sort == ascending sort of ~bits; label in LSB
                const unsigned key = ((~__float_as_uint(e)) & 0xFFFFFFFEu) | l;
                keys[(size_t)segBase + (unsigned)vm] = key;
                if (l) {
                    const float seed = sigmoidf_(p4[vm]);
                    const float df   = seed - d;
                    localSeed += df * df;
                }
            }
        }
    }
    __shared__ float ssum;
    if (threadIdx.x == 0) ssum = 0.0f;
    __syncthreads();
    atomicAdd(&ssum, localSeed);
    __syncthreads();
    if (threadIdx.x == 0) atomicAdd(&seedfg[b], ssum);
}

// ----------------------------------------------------------- radix sort -----
__global__ void k_hist(const unsigned* __restrict__ src, unsigned* __restrict__ gHist,
                       int shift) {
    const int seg = blockIdx.y, blk = blockIdx.x;
    __shared__ unsigned h[256];
    h[threadIdx.x] = 0;
    __syncthreads();
    const size_t base = ((size_t)seg << 20) + (size_t)blk * STILE;
    for (int i = threadIdx.x; i < STILE; i += 256) {
        const unsigned d = (src[base + i] >> shift) & 255u;
        atomicAdd(&h[d], 1u);
    }
    __syncthreads();
    gHist[((size_t)(seg * 256 + threadIdx.x)) * SBLKS + blk] = h[threadIdx.x];
}

__global__ void k_scan(unsigned* __restrict__ gHist) {
    const int seg = blockIdx.x;
    const int d   = threadIdx.x;
    unsigned* p = gHist + ((size_t)(seg * 256 + d)) * SBLKS;
    unsigned run = 0;
    for (int b = 0; b < SBLKS; ++b) { const unsigned v = p[b]; p[b] = run; run += v; }
    __shared__ unsigned tot[256];
    __shared__ unsigned start[256];
    tot[d] = run;
    __syncthreads();
    if (d == 0) {
        unsigned s = 0;
        for (int i = 0; i < 256; ++i) { start[i] = s; s += tot[i]; }
    }
    __syncthreads();
    const unsigned s0 = start[d];
    for (int b = 0; b < SBLKS; ++b) p[b] += s0;
}

__global__ void __launch_bounds__(256)
k_scatter(const unsigned* __restrict__ src, unsigned* __restrict__ dst,
          const unsigned* __restrict__ gHist, int shift) {
    const int seg = blockIdx.y, blk = blockIdx.x;
    const int tid = threadIdx.x, lane = tid & 31, wid = tid >> 5;
    __shared__ unsigned wh[8 * 256];
    __shared__ unsigned runBase[256];
    runBase[tid] = 0;
    const size_t sbase = ((size_t)seg << 20) + (size_t)blk * STILE;
    unsigned* dseg = dst + ((size_t)seg << 20);
    __syncthreads();
    for (int r = 0; r < STILE / 256; ++r) {
        for (int i = tid; i < 8 * 256; i += 256) wh[i] = 0;
        __syncthreads();
        const unsigned key = src[sbase + r * 256 + tid];
        const unsigned dig = (key >> shift) & 255u;
        // wave32 ballot-match: find peers with equal digit (stable by lane order)
        unsigned peers = 0xFFFFFFFFu;
#pragma unroll
        for (int bb = 0; bb < 8; ++bb) {
            const unsigned vote = (unsigned)__ballot(((dig >> bb) & 1u) != 0u);
            peers &= ((dig >> bb) & 1u) ? vote : ~vote;
        }
        const unsigned rank = __popc(peers & ((1u << lane) - 1u));
        if (rank == 0) wh[wid * 256 + dig] = __popc(peers);
        __syncthreads();
        unsigned cross = 0;
        for (int w = 0; w < wid; ++w) cross += wh[w * 256 + dig];
        const unsigned base = runBase[dig];
        const unsigned goff = gHist[((size_t)(seg * 256 + dig)) * SBLKS + blk];
        dseg[goff + base + cross + rank] = key;
        __syncthreads();
        unsigned tsum = 0;
#pragma unroll
        for (int w = 0; w < 8; ++w) tsum += wh[w * 256 + tid];
        runBase[tid] += tsum;
        __syncthreads();
    }
}

// ----------------------------------------------------------- ones scan ------
__global__ void k_lcnt(const unsigned* __restrict__ keys, unsigned* __restrict__ lCnt) {
    const int seg = blockIdx.y, blk = blockIdx.x;
    const size_t base = ((size_t)seg << 20) + (size_t)blk * STILE;
    unsigned c = 0;
    for (int i = threadIdx.x; i < STILE; i += 256) c += keys[base + i] & 1u;
    __shared__ unsigned s;
    if (threadIdx.x == 0) s = 0;
    __syncthreads();
    atomicAdd(&s, c);
    __syncthreads();
    if (threadIdx.x == 0) lCnt[seg * SBLKS + blk] = s;
}

__global__ void k_lscan(unsigned* __restrict__ lCnt, unsigned* __restrict__ segOnes) {
    const int seg = threadIdx.x;
    if (seg < NSEGS) {
        unsigned* p = lCnt + seg * SBLKS;
        unsigned run = 0;
        for (int b = 0; b < SBLKS; ++b) { const unsigned v = p[b]; p[b] = run; run += v; }
        segOnes[seg] = run;
    }
}

// ------------------------------------------------------------- lovasz -------
__global__ void __launch_bounds__(256)
k_lovasz(const unsigned* __restrict__ keys, const unsigned* __restrict__ lOff,
         const unsigned* __restrict__ segOnes, float* __restrict__ instLoss) {
    const int seg = blockIdx.y, blk = blockIdx.x;
    const int tid = threadIdx.x, lane = tid & 31, wid = tid >> 5;
    __shared__ unsigned waveOnes[8];
    __shared__ unsigned runOnes;
    __shared__ float bsum;
    if (tid == 0) { runOnes = 0; bsum = 0.0f; }
    __syncthreads();
    const unsigned blkBase = lOff[seg * SBLKS + blk];
    const float gts = (float)segOnes[seg];
    const size_t base = ((size_t)seg << 20) + (size_t)blk * STILE;
    float local = 0.0f;
    for (int r = 0; r < STILE / 256; ++r) {
        const unsigned key = keys[base + r * 256 + tid];
        const unsigned l   = key & 1u;
        const unsigned om  = (unsigned)__ballot(l != 0u);
        const unsigned incl = __popc(om & ((2u << lane) - 1u));  // inclusive
        if (lane == 0) waveOnes[wid] = __popc(om);
        __syncthreads();
        unsigned cross = 0;
        for (int w = 0; w < wid; ++w) cross += waveOnes[w];
        const unsigned rr = blkBase + runOnes + cross + incl;    // ones <= i
        const int i = blk * STILE + r * 256 + tid;               // seg-local pos
        const float e  = __uint_as_float(~key);
        const float r1 = (float)rr;
        const float uni = gts + ((float)(i + 1) - r1);
        const float jac = 1.0f - (gts - r1) * fast_rcp(uni);
        float jprev;
        if (i == 0) {
            jprev = 0.0f;                                        // grad_0 = jac_0
        } else {
            const float r0 = r1 - (float)l;
            const float u0 = gts + ((float)i - r0);
            jprev = 1.0f - (gts - r0) * fast_rcp(u0);
        }
        local += e * (jac - jprev);
        __syncthreads();
        if (tid == 0) {
            unsigned t = 0;
            for (int w = 0; w < 8; ++w) t += waveOnes[w];
            runOnes += t;
        }
        __syncthreads();
    }
    atomicAdd(&bsum, local);
    __syncthreads();
    if (tid == 0) atomicAdd(&instLoss[seg], bsum);
}

// -------------------------------------------------------------- final -------
__global__ void k_final(const float* __restrict__ stats, const float* __restrict__ derived,
                        const float* __restrict__ obj, const float* __restrict__ instLoss,
                        const float* __restrict__ seedfg, float* __restrict__ out) {
    if (threadIdx.x == 0 && blockIdx.x == 0) {
        float tot = 0.0f;
        for (int b = 0; b < 2; ++b) {
            const float so = fmaxf(obj[b], 1.0f);
            float inst = 0.0f, var = 0.0f;
            for (int k = 0; k < 8; ++k) {
                const float* dp = derived + (b * 8 + k) * 8;
                inst += instLoss[b * 8 + k] * dp[5];
                var  += dp[6] * dp[5];
            }
            inst /= so;
            var  /= so;
            const float seed = (stats[96 + b] + seedfg[b]) / 131072.0f;  // H*W
            tot += inst + 10.0f * var + seed;
        }
        out[0] = tot * 0.5f;   // mean over B=2
    }
}

// ------------------------------------------------------------- launch -------
extern "C" void kernel_launch(void* const* d_in, const int* in_sizes, int n_in,
                              void* d_out, int out_size, void* d_ws, size_t ws_size,
                              hipStream_t stream) {
    (void)in_sizes; (void)n_in; (void)out_size; (void)ws_size;
    const float* pred = (const float*)d_in[0];
    const int*   inst = (const int*)d_in[1];
    float*       out  = (float*)d_out;

    char* ws = (char*)d_ws;
    unsigned* keys0   = (unsigned*)ws;                                   // 64 MB
    unsigned* keys1   = (unsigned*)(ws + (1ull << 26));                  // 64 MB
    unsigned* gHist   = (unsigned*)(ws + (1ull << 27));                  // 4 MB
    unsigned* lCnt    = (unsigned*)(ws + (1ull << 27) + (4ull << 20));   // 16 KB
    unsigned* segOnes = lCnt + NSEGS * SBLKS;                            // 64 B
    float* stats    = (float*)(segOnes + NSEGS);  // 98 floats (incl seed_bg[2])
    float* derived  = stats + 98;                 // 128 floats
    float* obj      = derived + 128;              // 2
    float* instLoss = obj + 2;                    // 16
    float* seedfg   = instLoss + 16;              // 2

    k_init<<<1, 256, 0, stream>>>(stats);
    k_stats<<<2048, 256, 0, stream>>>(pred, inst, stats);
    k_finalize<<<1, 32, 0, stream>>>(stats, derived, obj);
    k_distkeys<<<dim3(512, 2), 256, 0, stream>>>(pred, inst, derived, keys0, seedfg);

    for (int p = 0; p < 4; ++p) {
        const unsigned* src = (p & 1) ? keys1 : keys0;
        unsigned*       dst = (p & 1) ? keys0 : keys1;
        k_hist<<<dim3(SBLKS, NSEGS), 256, 0, stream>>>(src, gHist, p * 8);
        k_scan<<<NSEGS, 256, 0, stream>>>(gHist);
        k_scatter<<<dim3(SBLKS, NSEGS), 256, 0, stream>>>(src, dst, gHist, p * 8);
    }
    // 4 passes end back in keys0

    k_lcnt<<<dim3(SBLKS, NSEGS), 256, 0, stream>>>(keys0, lCnt);
    k_lscan<<<1, 16, 0, stream>>>(lCnt, segOnes);
    k_lovasz<<<dim3(SBLKS, NSEGS), 256, 0, stream>>>(keys0, lCnt, segOnes, instLoss);
    k_final<<<1, 1, 0, stream>>>(stats, derived, obj, instLoss, seedfg, out);
}